// NewBottleneck_4569845203617
// MI455X (gfx1250) — compile-verified
//
#include <hip/hip_runtime.h>
#include <math.h>

typedef float v2f __attribute__((ext_vector_type(2)));
typedef float v4f __attribute__((ext_vector_type(4)));
typedef float v8f __attribute__((ext_vector_type(8)));

#define HWSZ   3136
#define CIN    256
#define BD     32
#define COUT   256
#define BATCH  32
#define ACT    18
#define NSLABS 28            // 3136 / 112
#define NTILE  112           // columns per workgroup (7 waves * 16)
#define EPS    1e-5f
#define NSTAT  (BATCH * HWSZ)   // 100352 samples per BN1 channel

// workspace layout (float offsets)
#define WS_SUM    0          // 32 floats
#define WS_SQ     32         // 32 floats
#define WS_LF     64         // latent_flat: 1024 floats
#define WS_FINAL  1088       // final per-(b,c): 8192 floats
#define WS_LATENT 9280       // latent: 32*32*3136 = 3,211,264 floats (~12.85 MB)

// ---------------------------------------------------------------------------
// 0) zero BN1 stats accumulators (ws is poisoned by harness)
// ---------------------------------------------------------------------------
__global__ void k_zero(float* __restrict__ p) { p[threadIdx.x] = 0.0f; }

// ---------------------------------------------------------------------------
// 1) compress GEMM: latent[b,o,hw] = sum_c Wc[o,c] * F[b,c,hw]
//    per-WG: one image b, 112-column slab. 7 waves, each owns 16 columns and
//    computes both 16-row M tiles (BD=32) via v_wmma_f32_16x16x4_f32.
//    Feature loads are NON-TEMPORAL (read-once stream; keep L2 for latent).
//    Fused: per-channel sum / sumsq for BatchNorm1 (training stats).
// ---------------------------------------------------------------------------
__global__ __launch_bounds__(224)
void k_compress(const float* __restrict__ F, const float* __restrict__ Wc,
                float* __restrict__ latent, float* __restrict__ gSum,
                float* __restrict__ gSq)
{
    // A staged in LDS as (k, k+1) float2 pairs, swizzled [p=k/2][m] so that
    // lanes 0-15 (varying m) hit consecutive banks: conflict-free ds_load_b64.
    __shared__ v2f   As[128 * 32];   // 32 KB
    __shared__ float sSum[BD];
    __shared__ float sSq[BD];

    const int tid  = threadIdx.x;
    const int b    = blockIdx.x / NSLABS;
    const int slab = blockIdx.x % NSLABS;

    // cooperative load of Wc (coalesced float2 reads along k)
    for (int i = tid; i < 128 * 32; i += 224) {
        const int m = i >> 7;          // 0..31
        const int p = i & 127;         // 0..127 (k pair)
        As[p * 32 + m] = *(const v2f*)(Wc + m * CIN + 2 * p);
    }
    if (tid < BD) { sSum[tid] = 0.0f; sSq[tid] = 0.0f; }
    __syncthreads();

    const int wave = tid >> 5;
    const int lane = tid & 31;
    const int hi   = lane >> 4;        // which K half (per 16x16x4 f32 layout)
    const int ln   = lane & 15;        // M (for A) / N (for B) within tile
    const int n    = slab * NTILE + wave * 16 + ln;   // hw column

    const float* __restrict__ Fb = F + (size_t)b * CIN * HWSZ;

    v8f acc0 = {0.f, 0.f, 0.f, 0.f, 0.f, 0.f, 0.f, 0.f};  // rows 0..15
    v8f acc1 = {0.f, 0.f, 0.f, 0.f, 0.f, 0.f, 0.f, 0.f};  // rows 16..31

#pragma unroll 4
    for (int k0 = 0; k0 < CIN; k0 += 4) {
        const int kb = k0 + hi * 2;    // this lane's K pair base
        v2f bf;
        bf.x = __builtin_nontemporal_load(Fb + (size_t)kb * HWSZ + n);
        bf.y = __builtin_nontemporal_load(Fb + (size_t)(kb + 1) * HWSZ + n);
        const v2f a0 = As[(kb >> 1) * 32 + ln];        // M tile 0
        const v2f a1 = As[(kb >> 1) * 32 + 16 + ln];   // M tile 1
        acc0 = __builtin_amdgcn_wmma_f32_16x16x4_f32(
                   false, a0, false, bf, (short)0, acc0, false, false);
        acc1 = __builtin_amdgcn_wmma_f32_16x16x4_f32(
                   false, a1, false, bf, (short)0, acc1, false, false);
    }

    // store latent (regular/RT: we want it L2-resident for k_latflat)
    // + fused BN1 stats (shfl-xor half-wave reduce -> LDS -> global)
    float* __restrict__ Lb = latent + (size_t)b * BD * HWSZ;
#pragma unroll
    for (int v = 0; v < 8; ++v) {
        const int o0 = v + 8 * hi;         // channel of acc0[v]
        const int o1 = 16 + v + 8 * hi;    // channel of acc1[v]
        const float x0 = acc0[v];
        const float x1 = acc1[v];
        Lb[(size_t)o0 * HWSZ + n] = x0;
        Lb[(size_t)o1 * HWSZ + n] = x1;

        float s0 = x0, q0 = x0 * x0, s1 = x1, q1 = x1 * x1;
        s0 += __shfl_xor(s0, 1);  q0 += __shfl_xor(q0, 1);
        s1 += __shfl_xor(s1, 1);  q1 += __shfl_xor(q1, 1);
        s0 += __shfl_xor(s0, 2);  q0 += __shfl_xor(q0, 2);
        s1 += __shfl_xor(s1, 2);  q1 += __shfl_xor(q1, 2);
        s0 += __shfl_xor(s0, 4);  q0 += __shfl_xor(q0, 4);
        s1 += __shfl_xor(s1, 4);  q1 += __shfl_xor(q1, 4);
        s0 += __shfl_xor(s0, 8);  q0 += __shfl_xor(q0, 8);
        s1 += __shfl_xor(s1, 8);  q1 += __shfl_xor(q1, 8);
        if (ln == 0) {
            atomicAdd(&sSum[o0], s0);  atomicAdd(&sSq[o0], q0);
            atomicAdd(&sSum[o1], s1);  atomicAdd(&sSq[o1], q1);
        }
    }
    __syncthreads();
    if (tid < BD) {
        atomicAdd(&gSum[tid], sSum[tid]);
        atomicAdd(&gSq[tid],  sSq[tid]);
    }
}

// ---------------------------------------------------------------------------
// 2) latent_flat[b,o] = mean_hw relu(bn1(latent[b,o,hw]))   (L2-resident read)
// ---------------------------------------------------------------------------
__global__ __launch_bounds__(256)
void k_latflat(const float* __restrict__ latent, const float* __restrict__ gSum,
               const float* __restrict__ gSq, const float* __restrict__ g1,
               const float* __restrict__ b1, float* __restrict__ lf)
{
    __shared__ float red[256];
    const int o = blockIdx.x & 31;
    const int b = blockIdx.x >> 5;

    const float mean = gSum[o] * (1.0f / NSTAT);
    const float var  = gSq[o] * (1.0f / NSTAT) - mean * mean;
    const float inv  = rsqrtf(var + EPS);
    const float sc   = inv * g1[o];
    const float sh   = b1[o] - mean * sc;

    const float* __restrict__ L = latent + ((size_t)b * BD + o) * HWSZ;
    float s = 0.0f;
    for (int i = threadIdx.x; i < HWSZ; i += 256) {
        s += fmaxf(L[i] * sc + sh, 0.0f);
    }
    red[threadIdx.x] = s;
    __syncthreads();
    for (int st = 128; st > 0; st >>= 1) {
        if (threadIdx.x < st) red[threadIdx.x] += red[threadIdx.x + st];
        __syncthreads();
    }
    if (threadIdx.x == 0) lf[b * BD + o] = red[0] * (1.0f / HWSZ);
}

// ---------------------------------------------------------------------------
// 3) tiny head: action embed, combine, expand, BN2 (+relu) -> final[b,c]
//    Output is spatially constant, so BN2 stats over (B,H,W) == stats over B.
// ---------------------------------------------------------------------------
__global__ __launch_bounds__(256)
void k_head(const float* __restrict__ lf, const float* __restrict__ actions,
            const float* __restrict__ Wa, const float* __restrict__ Wk,
            const float* __restrict__ We, const float* __restrict__ g2,
            const float* __restrict__ b2, float* __restrict__ finalv)
{
    __shared__ float comb[BATCH][2 * BD];   // 8 KB
    __shared__ float c2[BATCH][BD];         // 4 KB
    __shared__ float ov[BATCH][COUT];       // 32 KB
    const int tid = threadIdx.x;

    for (int i = tid; i < BATCH * BD; i += 256) {
        const int b = i >> 5, o = i & 31;
        comb[b][o] = lf[i];
        float s = 0.0f;
        for (int j = 0; j < ACT; ++j) s += actions[b * ACT + j] * Wa[o * ACT + j];
        comb[b][BD + o] = s;
    }
    __syncthreads();
    for (int i = tid; i < BATCH * BD; i += 256) {
        const int b = i >> 5, o = i & 31;
        float s = 0.0f;
        for (int j = 0; j < 2 * BD; ++j) s += comb[b][j] * Wk[o * 2 * BD + j];
        c2[b][o] = fmaxf(s, 0.0f);
    }
    __syncthreads();
    for (int i = tid; i < BATCH * COUT; i += 256) {
        const int b = i >> 8, co = i & 255;
        float s = 0.0f;
        for (int j = 0; j < BD; ++j) s += c2[b][j] * We[co * BD + j];
        ov[b][co] = s;
    }
    __syncthreads();
    {
        const int co = tid;     // 256 threads == 256 channels
        float m = 0.0f;
        for (int b = 0; b < BATCH; ++b) m += ov[b][co];
        m *= (1.0f / BATCH);
        float v = 0.0f;
        for (int b = 0; b < BATCH; ++b) { const float d = ov[b][co] - m; v += d * d; }
        v *= (1.0f / BATCH);
        const float inv = rsqrtf(v + EPS);
        const float sc  = inv * g2[co];
        const float sh  = b2[co] - m * sc;
        for (int b = 0; b < BATCH; ++b)
            finalv[b * COUT + co] = fmaxf(ov[b][co] * sc + sh, 0.0f);
    }
}

// ---------------------------------------------------------------------------
// 4) broadcast store: out[b,c,:] = final[b,c]
//    NON-TEMPORAL 128-bit stores: 102.8 MB written once, never re-read.
// ---------------------------------------------------------------------------
__global__ __launch_bounds__(256)
void k_bcast(const float* __restrict__ finalv, float* __restrict__ out)
{
    const float v = finalv[blockIdx.x];
    const v4f   q = {v, v, v, v};
    v4f* __restrict__ base = (v4f*)(out + (size_t)blockIdx.x * HWSZ);
    for (int i = threadIdx.x; i < HWSZ / 4; i += 256)
        __builtin_nontemporal_store(q, base + i);
}

// ---------------------------------------------------------------------------
extern "C" void kernel_launch(void* const* d_in, const int* in_sizes, int n_in,
                              void* d_out, int out_size, void* d_ws, size_t ws_size,
                              hipStream_t stream)
{
    (void)in_sizes; (void)n_in; (void)out_size; (void)ws_size;
    const float* F       = (const float*)d_in[0];
    const float* actions = (const float*)d_in[1];
    const float* Wc      = (const float*)d_in[2];
    const float* g1      = (const float*)d_in[3];
    const float* b1      = (const float*)d_in[4];
    const float* Wa      = (const float*)d_in[5];
    const float* Wk      = (const float*)d_in[6];
    const float* We      = (const float*)d_in[7];
    const float* g2      = (const float*)d_in[8];
    const float* b2      = (const float*)d_in[9];

    float* ws     = (float*)d_ws;
    float* gSum   = ws + WS_SUM;
    float* gSq    = ws + WS_SQ;
    float* lf     = ws + WS_LF;
    float* finalv = ws + WS_FINAL;
    float* latent = ws + WS_LATENT;
    float* out    = (float*)d_out;

    k_zero<<<1, 64, 0, stream>>>(gSum);                                  // sum+sq
    k_compress<<<BATCH * NSLABS, 224, 0, stream>>>(F, Wc, latent, gSum, gSq);
    k_latflat<<<BATCH * BD, 256, 0, stream>>>(latent, gSum, gSq, g1, b1, lf);
    k_head<<<1, 256, 0, stream>>>(lf, actions, Wa, Wk, We, g2, b2, finalv);
    k_bcast<<<BATCH * COUT, 256, 0, stream>>>(finalv, out);
}